// NodeModel_70153995813296
// MI455X (gfx1250) — compile-verified
//
#include <hip/hip_runtime.h>

typedef __attribute__((ext_vector_type(2))) float v2f;
typedef __attribute__((ext_vector_type(8))) float v8f;
typedef __attribute__((ext_vector_type(4))) int   v4i;

constexpr int N_NODES   = 100000;
constexpr int N_EDGES   = 1600000;
constexpr int D         = 32;    // D_NODE = D_EDGE = D_U
constexpr int OUT_DIM   = 64;
constexpr int K_DIM     = 128;   // IN1 = IN2 = 128
constexpr int ROW_STRIDE = 132;  // padded LDS row stride in floats (132 % 64 = 4 -> conflict-free)
constexpr int WAVES_PER_BLOCK = 8;
constexpr int A_TILE_FLOATS = 16 * ROW_STRIDE;   // per-wave A tile (16 rows x 128 cols, padded)
constexpr int W_TILE_FLOATS = 64 * ROW_STRIDE;   // W^T tile (64 rows x 128 cols, padded)
// LDS map (floats): [ wT | 8 x A-tile | 8 x 16 dest indices (as int) ]
constexpr int IDX_BASE_FLOATS = W_TILE_FLOATS + WAVES_PER_BLOCK * A_TILE_FLOATS;
constexpr int SMEM_FLOATS     = IDX_BASE_FLOATS + WAVES_PER_BLOCK * 16;

// ---------------------------------------------------------------------------
// Async global->LDS staging (CDNA5 GLOBAL_LOAD_ASYNC_TO_LDS_B128, ASYNCcnt),
// with a safe fallback to load+ds_store if the toolchain lacks the builtins.
// Probe-discovered signature: (v4i addrspace(1)* src, v4i addrspace(3)* dst,
//                              imm offset, imm cpol)
// ---------------------------------------------------------------------------
#if defined(__has_builtin)
#if __has_builtin(__builtin_amdgcn_global_load_async_to_lds_b128) && \
    __has_builtin(__builtin_amdgcn_s_wait_asynccnt)
#define USE_ASYNC_LDS 1
#endif
#endif
#ifndef USE_ASYNC_LDS
#define USE_ASYNC_LDS 0
#endif

typedef __attribute__((address_space(1))) v4i g_v4i;
typedef __attribute__((address_space(3))) v4i l_v4i;

__device__ __forceinline__ void copy16_b128(float* __restrict__ lds,
                                            const float* __restrict__ g)
{
#if USE_ASYNC_LDS
    __builtin_amdgcn_global_load_async_to_lds_b128(
        (g_v4i*)g, (l_v4i*)lds, /*imm offset*/ 0, /*cpol*/ 0);
#else
    *(float4*)lds = *(const float4*)g;
#endif
}

__device__ __forceinline__ void staging_commit()
{
#if USE_ASYNC_LDS
    asm volatile("" ::: "memory");
    __builtin_amdgcn_s_wait_asynccnt(0);
    asm volatile("" ::: "memory");
#endif
}

// ---------------------------------------------------------------------------
// Inner WMMA machinery: A(16x128) @ W(128x64) + bias into 4 tiles of v8f via
// V_WMMA_F32_16X16X4_F32 (fp32 XDL path, exact fp32 numerics).
//
// fp32 WMMA layouts (ISA 7.12.2):
//   A 16x4 : lane<16 -> (M=lane,   K=k+0..1), lane>=16 -> (M=lane-16, K=k+2..3)
//   B 4x16 : lane<16 -> (K=k+0..1, N=lane),   lane>=16 -> (K=k+2..3,  N=lane-16)
//   C/D    : vgpr v  -> row v + 8*(lane>>4), col = lane & 15
// ---------------------------------------------------------------------------
__device__ __forceinline__ void wmma_tile_f32(const float* __restrict__ aT,
                                              const float* __restrict__ wT,
                                              const float* __restrict__ bias,
                                              int lane, v8f acc[4])
{
    const int r    = lane & 15;
    const int koff = 2 * (lane >> 4);

    #pragma unroll
    for (int t = 0; t < 4; ++t) {
        const float bv = bias[t * 16 + r];
        #pragma unroll
        for (int v = 0; v < 8; ++v) acc[t][v] = bv;
    }

    #pragma unroll 4
    for (int kk = 0; kk < K_DIM / 4; ++kk) {
        const int k = kk * 4 + koff;
        v2f a = *(const v2f*)(aT + r * ROW_STRIDE + k);
        #pragma unroll
        for (int t = 0; t < 4; ++t) {
            v2f b = *(const v2f*)(wT + (t * 16 + r) * ROW_STRIDE + k);
            acc[t] = __builtin_amdgcn_wmma_f32_16x16x4_f32(
                false, a, false, b, (short)0, acc[t], false, false);
        }
    }
}

// Stage W^T (wT[n*132 + k] = W[k*64 + n]) into LDS with the whole block.
__device__ __forceinline__ void stage_wT(float* __restrict__ wT,
                                         const float* __restrict__ W)
{
    for (int i = threadIdx.x; i < OUT_DIM * K_DIM; i += blockDim.x) {
        const int n = i >> 7;     // / 128
        const int k = i & 127;    // % 128
        wT[n * ROW_STRIDE + k] = W[k * OUT_DIM + n];
    }
    __syncthreads();
}

// ---------------------------------------------------------------------------
// Kernel 1: edge MLP + scatter-add.
//   m_in = [x[dest] | x[src] | edge_attr | u[batch[src]]]  (128)
//   msg  = relu(m_in @ W1 + b1)                            (64)
//   agg[dest] += msg  (global_atomic_add_f32)
// One wave = 16 edges x all 64 output columns.
// ---------------------------------------------------------------------------
__global__ void edge_mlp_scatter(const float* __restrict__ x,
                                 const int*   __restrict__ edge_index,  // [2][N_EDGES]
                                 const float* __restrict__ edge_attr,
                                 const float* __restrict__ u,
                                 const int*   __restrict__ batch,
                                 const float* __restrict__ W1,          // [128][64]
                                 const float* __restrict__ b1,          // [64]
                                 float*       __restrict__ agg,         // [N_NODES][64]
                                 int n_groups)
{
    extern __shared__ float smem[];
    float* wT = smem;
    const int lane = threadIdx.x & 31;
    const int wave = threadIdx.x >> 5;
    float* aT  = smem + W_TILE_FLOATS + wave * A_TILE_FLOATS;
    int*   dIx = (int*)(smem + IDX_BASE_FLOATS) + wave * 16;

    stage_wT(wT, W1);   // block-wide, then barrier

    const int group = blockIdx.x * WAVES_PER_BLOCK + wave;
    if (group >= n_groups) return;             // wave-uniform: EXEC all-ones below

    const int base = group * 16;
    const int half = lane >> 4;
    const int r    = lane & 15;
    const int e    = base + r;

    // Prefetch next group's index block while we work on this one.
    if (group + WAVES_PER_BLOCK < n_groups) {
        __builtin_prefetch(edge_index + base + 16, 0, 1);
        __builtin_prefetch(edge_index + N_EDGES + base + 16, 0, 1);
    }

    // Gather one half-row (64 floats) per lane into the wave's A tile.
    const long srcI  = (long)edge_index[e];
    const long destI = (long)edge_index[N_EDGES + e];
    if (half == 0) dIx[r] = (int)destI;        // stash dest indices for epilogue
    const float* s0;
    const float* s1;
    if (half == 0) {
        s0 = x + destI * D;
        s1 = x + srcI  * D;
    } else {
        s0 = edge_attr + (long)e * D;
        s1 = u + (long)batch[srcI] * D;
    }
    float* dstRow = aT + r * ROW_STRIDE + half * 64;
    #pragma unroll
    for (int i = 0; i < 8; ++i) copy16_b128(dstRow + 4 * i,      s0 + 4 * i);
    #pragma unroll
    for (int i = 0; i < 8; ++i) copy16_b128(dstRow + 32 + 4 * i, s1 + 4 * i);
    staging_commit();   // s_wait_asynccnt 0 (async path); DS is wave-in-order

    v8f acc[4];
    wmma_tile_f32(aT, wT, b1, lane, acc);

    // ReLU + scatter-add to destination nodes (dest indices from LDS stash).
    #pragma unroll
    for (int v = 0; v < 8; ++v) {
        const long dn = (long)dIx[v + 8 * half];
        float* dst = agg + dn * OUT_DIM + r;
        #pragma unroll
        for (int t = 0; t < 4; ++t) {
            float val = acc[t][v];
            val = val > 0.0f ? val : 0.0f;
            atomicAdd(dst + t * 16, val);
        }
    }
}

// ---------------------------------------------------------------------------
// Kernel 2: node MLP.
//   n_in = [x | agg | u[batch]]  (128) ;  out = relu(n_in @ W2 + b2)  (64)
// ---------------------------------------------------------------------------
__global__ void node_mlp(const float* __restrict__ x,
                         const float* __restrict__ agg,
                         const float* __restrict__ u,
                         const int*   __restrict__ batch,
                         const float* __restrict__ W2,   // [128][64]
                         const float* __restrict__ b2,   // [64]
                         float*       __restrict__ out,  // [N_NODES][64]
                         int n_groups)
{
    extern __shared__ float smem[];
    float* wT = smem;
    const int lane = threadIdx.x & 31;
    const int wave = threadIdx.x >> 5;
    float* aT = smem + W_TILE_FLOATS + wave * A_TILE_FLOATS;

    stage_wT(wT, W2);

    const int group = blockIdx.x * WAVES_PER_BLOCK + wave;
    if (group >= n_groups) return;

    const int base = group * 16;
    const int half = lane >> 4;
    const int r    = lane & 15;
    const long i   = base + r;

    const float* s0;
    const float* s1;
    if (half == 0) {
        s0 = x + i * D;               // cols 0..31
        s1 = agg + i * OUT_DIM;       // cols 32..63
    } else {
        s0 = agg + i * OUT_DIM + 32;  // cols 64..95
        s1 = u + (long)batch[i] * D;  // cols 96..127
    }
    float* dstRow = aT + r * ROW_STRIDE + half * 64;
    #pragma unroll
    for (int k = 0; k < 8; ++k) copy16_b128(dstRow + 4 * k,      s0 + 4 * k);
    #pragma unroll
    for (int k = 0; k < 8; ++k) copy16_b128(dstRow + 32 + 4 * k, s1 + 4 * k);
    staging_commit();

    v8f acc[4];
    wmma_tile_f32(aT, wT, b2, lane, acc);

    #pragma unroll
    for (int v = 0; v < 8; ++v) {
        const long row = base + v + 8 * half;
        float* dst = out + row * OUT_DIM + r;
        #pragma unroll
        for (int t = 0; t < 4; ++t) {
            float val = acc[t][v];
            dst[t * 16] = val > 0.0f ? val : 0.0f;
        }
    }
}

// ---------------------------------------------------------------------------
extern "C" void kernel_launch(void* const* d_in, const int* in_sizes, int n_in,
                              void* d_out, int out_size, void* d_ws, size_t ws_size,
                              hipStream_t stream)
{
    (void)in_sizes; (void)n_in; (void)out_size; (void)ws_size;

    const float* x          = (const float*)d_in[0];
    const int*   edge_index = (const int*)  d_in[1];
    const float* edge_attr  = (const float*)d_in[2];
    const float* u          = (const float*)d_in[3];
    const int*   batch      = (const int*)  d_in[4];
    const float* W1         = (const float*)d_in[5];
    const float* b1         = (const float*)d_in[6];
    const float* W2         = (const float*)d_in[7];
    const float* b2         = (const float*)d_in[8];
    float* out = (float*)d_out;
    float* agg = (float*)d_ws;    // [N_NODES][64] scratch

    (void)hipMemsetAsync(agg, 0, (size_t)N_NODES * OUT_DIM * sizeof(float), stream);

    const size_t shmem = (size_t)SMEM_FLOATS * sizeof(float);

    const int egroups = N_EDGES / 16;   // 100000
    const int eblocks = (egroups + WAVES_PER_BLOCK - 1) / WAVES_PER_BLOCK;
    edge_mlp_scatter<<<eblocks, 32 * WAVES_PER_BLOCK, shmem, stream>>>(
        x, edge_index, edge_attr, u, batch, W1, b1, agg, egroups);

    const int ngroups = N_NODES / 16;   // 6250
    const int nblocks = (ngroups + WAVES_PER_BLOCK - 1) / WAVES_PER_BLOCK;
    node_mlp<<<nblocks, 32 * WAVES_PER_BLOCK, shmem, stream>>>(
        x, agg, u, batch, W2, b2, out, ngroups);
}